// GNNLayer_70875550319239
// MI455X (gfx1250) — compile-verified
//
#include <hip/hip_runtime.h>

// ---------------------------------------------------------------------------
// NNConv GNN layer, fused for MI455X (gfx1250, wave32, WMMA f32 16x16x4).
// msg[t] = sum_k e[t,k]*(v[src_t]@Wk) + v[src_t]@B   (never materialize [E,256])
// Edge kernel: one wave32 handles TILES_PER_WAVE consecutive 16-edge tiles,
// with the five constant 16x16 B matrices staged in VGPRs once per wave.
// Full-tile fast path keeps EXEC uniform (no per-row masking) in the hot loop.
// ---------------------------------------------------------------------------

typedef float v2f __attribute__((ext_vector_type(2)));
typedef float v8f __attribute__((ext_vector_type(8)));

#define EPS_BN   1e-5f
#define SLOPE_LR 0.01f
#define TILES_PER_WAVE 8

// ---- zero workspace (summed[N*16], cnt[N], stats[32]) ----------------------
__global__ void init_ws_kernel(float* __restrict__ ws, long n) {
    long i = (long)blockIdx.x * blockDim.x + threadIdx.x;
    long stride = (long)gridDim.x * blockDim.x;
    for (; i < n; i += stride) ws[i] = 0.0f;
}

// ---- in-degree histogram (scatter-mean denominator) ------------------------
__global__ void degree_kernel(const int* __restrict__ ei, float* __restrict__ cnt,
                              int E_) {
    int i = blockIdx.x * blockDim.x + threadIdx.x;
    int stride = gridDim.x * blockDim.x;
    for (; i < E_; i += stride)
        atomicAdd(&cnt[ei[E_ + i]], 1.0f);
}

// ---- edge kernel: one wave32 per TILES_PER_WAVE 16-edge tiles --------------
__global__ __launch_bounds__(256) void edge_kernel(
    const float* __restrict__ v, const float* __restrict__ e,
    const int*   __restrict__ ei, const float* __restrict__ enet_w,
    const float* __restrict__ enet_b,
    float* __restrict__ summed,
    int E_, int numTiles)
{
    int lane = threadIdx.x & 31;
    int wid  = blockIdx.x * (blockDim.x >> 5) + (threadIdx.x >> 5);
    int hw = lane >> 4;                             // half-wave select (K split)
    int lm = lane & 15;                             // M (A rows) / N=o (B cols)

    if (wid * TILES_PER_WAVE >= numTiles) return;   // wave-uniform

    // B-matrix registers (loaded once per wave): Wk (k=0..3) + bias matrix,
    // 4 K-chunks of 4 each.  B layout: lane -> N=lm; chunk c: rows i0, i0+1.
    v2f Bw[4][4];
    v2f Bb[4];
#pragma unroll
    for (int c = 0; c < 4; ++c) {
        int i0 = 4 * c + 2 * hw;
#pragma unroll
        for (int k = 0; k < 4; ++k) {
            Bw[k][c].x = enet_w[((i0    ) * 16 + lm) * 4 + k];
            Bw[k][c].y = enet_w[((i0 + 1) * 16 + lm) * 4 + k];
        }
        Bb[c].x = enet_b[(i0    ) * 16 + lm];
        Bb[c].y = enet_b[(i0 + 1) * 16 + lm];
    }

#pragma unroll 1
    for (int it = 0; it < TILES_PER_WAVE; ++it) {
        int tile = wid * TILES_PER_WAVE + it;
        if (tile >= numTiles) break;                // wave-uniform
        int tile_base = tile * 16;

        // A-matrix gather: row M=lm of the tile = v[src[tile_base+lm]]
        int t_row = tile_base + lm;
        int tr = t_row < E_ ? t_row : 0;            // clamp tail (masked later)
        const float* vrow = v + (long)ei[tr] * 16;  // src = edge_index[0][t]

        v8f acc[5];
#pragma unroll
        for (int m = 0; m < 5; ++m)
#pragma unroll
            for (int q = 0; q < 8; ++q) acc[m][q] = 0.0f;

#pragma unroll
        for (int c = 0; c < 4; ++c) {
            v2f a;
            a.x = vrow[4 * c + 2 * hw + 0];
            a.y = vrow[4 * c + 2 * hw + 1];
            acc[0] = __builtin_amdgcn_wmma_f32_16x16x4_f32(false, a, false, Bw[0][c], (short)0, acc[0], false, false);
            acc[1] = __builtin_amdgcn_wmma_f32_16x16x4_f32(false, a, false, Bw[1][c], (short)0, acc[1], false, false);
            acc[2] = __builtin_amdgcn_wmma_f32_16x16x4_f32(false, a, false, Bw[2][c], (short)0, acc[2], false, false);
            acc[3] = __builtin_amdgcn_wmma_f32_16x16x4_f32(false, a, false, Bw[3][c], (short)0, acc[3], false, false);
            acc[4] = __builtin_amdgcn_wmma_f32_16x16x4_f32(false, a, false, Bb[c],    (short)0, acc[4], false, false);
        }

        // Combine with per-edge e[t,k] in accumulator layout, scatter to dst.
        // C/D layout: VGPR j, lane -> (M = j + 8*hw, N = lm).
        if (tile_base + 16 <= E_) {
            // Fast path: full tile, no per-row masking, fully unrolled.
#pragma unroll
            for (int j = 0; j < 8; ++j) {
                int t = tile_base + j + 8 * hw;
                const float* erow = e + (long)t * 4;
                float msg = acc[4][j]
                          + erow[0] * acc[0][j] + erow[1] * acc[1][j]
                          + erow[2] * acc[2][j] + erow[3] * acc[3][j];
                int d = ei[E_ + t];                 // dst = edge_index[1][t]
                atomicAdd(&summed[(long)d * 16 + lm], msg);
            }
        } else {
            // Tail tile: per-row bounds check.
#pragma unroll
            for (int j = 0; j < 8; ++j) {
                int t = tile_base + j + 8 * hw;
                if (t < E_) {
                    const float* erow = e + (long)t * 4;
                    float msg = acc[4][j]
                              + erow[0] * acc[0][j] + erow[1] * acc[1][j]
                              + erow[2] * acc[2][j] + erow[3] * acc[3][j];
                    int d = ei[E_ + t];
                    atomicAdd(&summed[(long)d * 16 + lm], msg);
                }
            }
        }
    }
}

// ---- node kernel: scatter-mean + v@root + bias, accumulate BN stats --------
__global__ __launch_bounds__(256) void node_kernel(
    const float* __restrict__ v, const float* __restrict__ root,
    const float* __restrict__ bias,
    const float* __restrict__ summed, const float* __restrict__ cnt,
    float* __restrict__ out, float* __restrict__ stats,
    int N_, int numTiles)
{
    int lane = threadIdx.x & 31;
    int wid  = blockIdx.x * (blockDim.x >> 5) + (threadIdx.x >> 5);
    if (wid >= numTiles) return;
    int tile_base = wid * 16;
    int hw = lane >> 4;
    int lm = lane & 15;

    v2f Br[4];
#pragma unroll
    for (int c = 0; c < 4; ++c) {
        int i0 = 4 * c + 2 * hw;
        Br[c].x = root[(i0    ) * 16 + lm];
        Br[c].y = root[(i0 + 1) * 16 + lm];
    }

    int n_row = tile_base + lm;
    int nr = n_row < N_ ? n_row : 0;
    const float* vrow = v + (long)nr * 16;

    // Every accumulator element in lane l has N = lm -> init C with bias[lm].
    float b = bias[lm];
    v8f acc;
#pragma unroll
    for (int q = 0; q < 8; ++q) acc[q] = b;

#pragma unroll
    for (int c = 0; c < 4; ++c) {
        v2f a;
        a.x = vrow[4 * c + 2 * hw + 0];
        a.y = vrow[4 * c + 2 * hw + 1];
        acc = __builtin_amdgcn_wmma_f32_16x16x4_f32(false, a, false, Br[c], (short)0, acc, false, false);
    }

    float lsum = 0.0f, lsq = 0.0f;
#pragma unroll
    for (int j = 0; j < 8; ++j) {
        int n = tile_base + j + 8 * hw;
        if (n < N_) {
            float c   = cnt[n];
            float val = summed[(long)n * 16 + lm] / fmaxf(c, 1.0f) + acc[j];
            out[(long)n * 16 + lm] = val;
            lsum += val;
            lsq  += val * val;
        }
    }
    atomicAdd(&stats[lm],      lsum);
    atomicAdd(&stats[16 + lm], lsq);
}

// ---- fold batch stats into scale/shift -------------------------------------
__global__ void finalize_stats_kernel(const float* __restrict__ stats,
                                      const float* __restrict__ gamma,
                                      const float* __restrict__ beta,
                                      float* __restrict__ sc, float fN)
{
    int o = threadIdx.x;
    if (o < 16) {
        float mu  = stats[o] / fN;
        float var = stats[16 + o] / fN - mu * mu;
        float g   = gamma[o] * rsqrtf(var + EPS_BN);
        sc[o]      = g;                 // scale
        sc[16 + o] = beta[o] - g * mu;  // shift
    }
}

// ---- BN + LeakyReLU, in place on d_out -------------------------------------
__global__ void bn_lrelu_kernel(float* __restrict__ out,
                                const float* __restrict__ sc, long total)
{
    long i = (long)blockIdx.x * blockDim.x + threadIdx.x;
    long stride = (long)gridDim.x * blockDim.x;
    for (; i < total; i += stride) {
        int o = (int)(i & 15);
        float y = sc[o] * out[i] + sc[16 + o];
        out[i] = y >= 0.0f ? y : SLOPE_LR * y;
    }
}

// ---------------------------------------------------------------------------
extern "C" void kernel_launch(void* const* d_in, const int* in_sizes, int n_in,
                              void* d_out, int out_size, void* d_ws, size_t ws_size,
                              hipStream_t stream)
{
    const float* v      = (const float*)d_in[0];
    const float* e      = (const float*)d_in[1];
    const int*   ei     = (const int*)  d_in[2];
    const float* enet_w = (const float*)d_in[3];
    const float* enet_b = (const float*)d_in[4];
    const float* root   = (const float*)d_in[5];
    const float* bias   = (const float*)d_in[6];
    const float* gam    = (const float*)d_in[7];
    const float* bet    = (const float*)d_in[8];
    float* out = (float*)d_out;

    int N_ = in_sizes[0] / 16;   // v is [N, 16]
    int E_ = in_sizes[1] / 4;    // e is [E, 4]

    // workspace layout
    float* summed = (float*)d_ws;                 // N*16
    float* cnt    = summed + (long)N_ * 16;       // N
    float* stats  = cnt + N_;                     // 32 (sum, sumsq)
    float* sc     = stats + 32;                   // 32 (scale, shift)

    const int tb = 256;
    long initN = (long)N_ * 16 + N_ + 32;
    init_ws_kernel<<<(int)((initN + tb - 1) / tb), tb, 0, stream>>>(summed, initN);

    int dBlocks = (E_ + tb - 1) / tb;
    if (dBlocks > 4096) dBlocks = 4096;
    degree_kernel<<<dBlocks, tb, 0, stream>>>(ei, cnt, E_);

    const int wavesPerBlock = tb / 32;
    int eTiles  = (E_ + 15) / 16;
    int eWaves  = (eTiles + TILES_PER_WAVE - 1) / TILES_PER_WAVE;
    int eBlocks = (eWaves + wavesPerBlock - 1) / wavesPerBlock;
    edge_kernel<<<eBlocks, tb, 0, stream>>>(v, e, ei, enet_w, enet_b,
                                            summed, E_, eTiles);

    int nTiles  = (N_ + 15) / 16;
    int nBlocks = (nTiles + wavesPerBlock - 1) / wavesPerBlock;
    node_kernel<<<nBlocks, tb, 0, stream>>>(v, root, bias, summed, cnt,
                                            out, stats, N_, nTiles);

    finalize_stats_kernel<<<1, 32, 0, stream>>>(stats, gam, bet, sc, (float)N_);

    long total = (long)N_ * 16;
    int bnBlocks = (int)((total + tb - 1) / tb);
    if (bnBlocks > 65535) bnBlocks = 65535;
    bn_lrelu_kernel<<<bnBlocks, tb, 0, stream>>>(out, sc, total);
}